// MultiHeadGraphAttention_71184787964209
// MI455X (gfx1250) — compile-verified
//
#include <hip/hip_runtime.h>
#include <hip/hip_bf16.h>

#define B_ 8
#define N_ 4096
#define F_ 128
#define EMB_ 64
#define H_ 3

// LDS W layout: K-pair interleaved rows, padded so lanes 0-15 (kpair p) and
// lanes 16-31 (kpair p+1) hit disjoint bank halves on b64 loads.
#define KP_STRIDE (2 * EMB_ + 32)   // 160 floats; 160 % 64 == 32

typedef float v2f __attribute__((ext_vector_type(2)));
typedef float v4f __attribute__((ext_vector_type(4)));
typedef float v8f __attribute__((ext_vector_type(8)));

// ---------------------------------------------------------------------------
// Stage A: h[b,h,n,:] = X[b,n,:] @ W_node[h] + b_node[h]
// One wave computes a 16x64 tile with V_WMMA_F32_16X16X4_F32 over K=128.
// W_node[h] staged in LDS K-pair-interleaved: each B fragment = 1 ds_load_b64.
// ---------------------------------------------------------------------------
__global__ __launch_bounds__(128) void gat_stageA(
    const float* __restrict__ X, const float* __restrict__ Wn,
    const float* __restrict__ bn, float* __restrict__ Hf) {
  __shared__ float Wl[(F_ / 2) * KP_STRIDE];   // 64 * 160 * 4B = 40 KB
  const int tid  = threadIdx.x;
  const int lane = tid & 31;
  const int wave = tid >> 5;
  const int h = blockIdx.y, b = blockIdx.z;

  // Cooperative interleaved fill: Wl[kp*KP_STRIDE + 2n + (k&1)] = W[k][n]
  const float* Wh = Wn + (size_t)h * F_ * EMB_;
  for (int i = tid; i < (F_ / 2) * (EMB_ / 4); i += 128) {  // 1024 iters total
    const int kp = i >> 4;            // kpair index 0..63
    const int n4 = (i & 15) << 2;     // column group 0,4,...,60
    const float* src = Wh + (size_t)(2 * kp) * EMB_ + n4;
    v4f w0 = *(const v4f*)(src);          // row 2kp
    v4f w1 = *(const v4f*)(src + EMB_);   // row 2kp+1
    float* dst = Wl + kp * KP_STRIDE + 2 * n4;
    v4f lo = {w0.x, w1.x, w0.y, w1.y};
    v4f hi = {w0.z, w1.z, w0.w, w1.w};
    *(v4f*)(dst)     = lo;
    *(v4f*)(dst + 4) = hi;
  }
  __syncthreads();

  const int ln16  = lane & 15;
  const int khalf = (lane >> 4) << 1;          // lanes 16-31 hold K+2 halves
  const int row0  = (blockIdx.x * 4 + wave) * 16;
  const float* Xrow = X + ((size_t)b * N_ + row0 + ln16) * F_;

  v8f acc0 = {}, acc1 = {}, acc2 = {}, acc3 = {};
  for (int k0 = 0; k0 < F_; k0 += 4) {
    v2f a = *(const v2f*)(Xrow + k0 + khalf);   // A[M=row][K=k0+khalf .. +1]
    const float* wk = Wl + ((k0 + khalf) >> 1) * KP_STRIDE + 2 * ln16;
    v2f b0 = *(const v2f*)(wk);        // {W[k][n],   W[k+1][n]}   one ds_load_b64
    v2f b1 = *(const v2f*)(wk + 32);   // cols +16
    v2f b2 = *(const v2f*)(wk + 64);   // cols +32
    v2f b3 = *(const v2f*)(wk + 96);   // cols +48
    acc0 = __builtin_amdgcn_wmma_f32_16x16x4_f32(false, a, false, b0, (short)0, acc0, false, false);
    acc1 = __builtin_amdgcn_wmma_f32_16x16x4_f32(false, a, false, b1, (short)0, acc1, false, false);
    acc2 = __builtin_amdgcn_wmma_f32_16x16x4_f32(false, a, false, b2, (short)0, acc2, false, false);
    acc3 = __builtin_amdgcn_wmma_f32_16x16x4_f32(false, a, false, b3, (short)0, acc3, false, false);
  }

  // C layout: VGPR r, lanes 0-15 -> M=r, lanes 16-31 -> M=8+r, col = lane%16
  const int rowbase = row0 + ((lane >> 4) << 3);
  float* out = Hf + (((size_t)b * H_ + h) * N_) * EMB_;
  const float bias0 = bn[h * EMB_ + ln16];
  const float bias1 = bn[h * EMB_ + 16 + ln16];
  const float bias2 = bn[h * EMB_ + 32 + ln16];
  const float bias3 = bn[h * EMB_ + 48 + ln16];
#pragma unroll
  for (int r = 0; r < 8; ++r) {
    float* orow = out + (size_t)(rowbase + r) * EMB_ + ln16;
    orow[0]  = acc0[r] + bias0;
    orow[16] = acc1[r] + bias1;
    orow[32] = acc2[r] + bias2;
    orow[48] = acc3[r] + bias3;
  }
}

// ---------------------------------------------------------------------------
// Stage A2: attention coefficients for batch B-1 only (the only batch whose
// scores survive in the reference).  One wave per (h, n).
// ---------------------------------------------------------------------------
__global__ __launch_bounds__(256) void gat_att_coef(
    const float* __restrict__ Hf, const float* __restrict__ Wa,
    float* __restrict__ a_src, float* __restrict__ a_dst) {
  const int gw   = (int)((blockIdx.x * blockDim.x + threadIdx.x) >> 5);
  const int lane = threadIdx.x & 31;
  if (gw >= H_ * N_) return;
  const int h = gw / N_, n = gw % N_;
  const float* hv = Hf + (((size_t)(B_ - 1) * H_ + h) * N_ + n) * EMB_;
  const float* wa = Wa + (size_t)h * 2 * EMB_;
  const float x0 = hv[lane], x1 = hv[lane + 32];
  float ssrc = x0 * wa[lane]        + x1 * wa[lane + 32];
  float sdst = x0 * wa[EMB_ + lane] + x1 * wa[EMB_ + lane + 32];
#pragma unroll
  for (int off = 16; off > 0; off >>= 1) {
    ssrc += __shfl_xor(ssrc, off, 32);
    sdst += __shfl_xor(sdst, off, 32);
  }
  if (lane == 0) { a_src[gw] = ssrc; a_dst[gw] = sdst; }
}

__global__ void gat_zero(float* __restrict__ p, int n) {
  int i = blockIdx.x * blockDim.x + threadIdx.x;
  if (i < n) p[i] = 0.0f;
}

// ---------------------------------------------------------------------------
// Stage B: per-edge scores (leaky_relu -> clip -> exp) + segment sums by src.
// ---------------------------------------------------------------------------
__global__ void gat_edge_scores(const int* __restrict__ edges,
                                const float* __restrict__ a_src,
                                const float* __restrict__ a_dst,
                                const float* __restrict__ ba,
                                float* __restrict__ score,
                                float* __restrict__ seg, int E) {
  int e = blockIdx.x * blockDim.x + threadIdx.x;
  if (e >= E) return;
  const int s = edges[2 * e], d = edges[2 * e + 1];
#pragma unroll
  for (int h = 0; h < H_; ++h) {
    float v = a_src[h * N_ + s] + a_dst[h * N_ + d] + ba[h];
    v = (v > 0.0f) ? v : 0.2f * v;            // leaky_relu(0.2)
    v = fminf(fmaxf(v, -2.0f), 2.0f);         // clip
    float sc = expf(v);
    score[(size_t)h * E + e] = sc;
    atomicAdd(&seg[h * N_ + s], sc);
  }
}

// edges are sorted by src (np.argwhere row-major); diagonal guarantees every
// row is non-empty, so every row_ptr slot gets written.
__global__ void gat_rowptr(const int* __restrict__ edges, int* __restrict__ rp, int E) {
  int e = blockIdx.x * blockDim.x + threadIdx.x;
  if (e >= E) return;
  const int s = edges[2 * e];
  if (e == 0 || edges[2 * (e - 1)] != s) rp[s] = e;
  if (e == E - 1) rp[N_] = E;
}

// ---------------------------------------------------------------------------
// Stage C: out[b,n,h,:] = sum_{e in row n} (score[h,e]/seg[h,n]) * h[b,h,dst,:]
// One wave per (b, n): each lane owns 2 columns x 3 heads.
// ---------------------------------------------------------------------------
__global__ __launch_bounds__(256) void gat_aggregate(
    const int* __restrict__ edges, const int* __restrict__ rp,
    const float* __restrict__ score, const float* __restrict__ seg,
    const float* __restrict__ Hf, float* __restrict__ out, int E) {
  const int gw   = (int)((blockIdx.x * blockDim.x + threadIdx.x) >> 5);
  const int lane = threadIdx.x & 31;
  if (gw >= B_ * N_) return;
  const int b = gw / N_, n = gw % N_;
  const int e0 = rp[n], e1 = rp[n + 1];
  const float inv0 = 1.0f / seg[0 * N_ + n];
  const float inv1 = 1.0f / seg[1 * N_ + n];
  const float inv2 = 1.0f / seg[2 * N_ + n];
  const int col = lane * 2;
  const float* Hb = Hf + (size_t)b * H_ * N_ * EMB_;
  v2f acc0 = {}, acc1 = {}, acc2 = {};
  for (int e = e0; e < e1; ++e) {
    const int d  = edges[2 * e + 1];
    const float w0 = score[(size_t)0 * E + e] * inv0;
    const float w1 = score[(size_t)1 * E + e] * inv1;
    const float w2 = score[(size_t)2 * E + e] * inv2;
    v2f h0 = *(const v2f*)(Hb + ((size_t)0 * N_ + d) * EMB_ + col);
    v2f h1 = *(const v2f*)(Hb + ((size_t)1 * N_ + d) * EMB_ + col);
    v2f h2 = *(const v2f*)(Hb + ((size_t)2 * N_ + d) * EMB_ + col);
    acc0 += w0 * h0;
    acc1 += w1 * h1;
    acc2 += w2 * h2;
  }
  float* o = out + (size_t)(b * N_ + n) * H_ * EMB_;
  *(v2f*)(o + 0 * EMB_ + col) = acc0;
  *(v2f*)(o + 1 * EMB_ + col) = acc1;
  *(v2f*)(o + 2 * EMB_ + col) = acc2;
}

// ---------------------------------------------------------------------------
extern "C" void kernel_launch(void* const* d_in, const int* in_sizes, int n_in,
                              void* d_out, int out_size, void* d_ws, size_t ws_size,
                              hipStream_t stream) {
  const float* X  = (const float*)d_in[0];   // (B,N,F)
  const float* Wn = (const float*)d_in[1];   // (H,F,EMB)
  const float* bn = (const float*)d_in[2];   // (H,EMB)
  const float* Wa = (const float*)d_in[3];   // (H,2*EMB)
  const float* ba = (const float*)d_in[4];   // (H,)
  const int* edges = (const int*)d_in[5];    // (E,2)
  const int E = in_sizes[5] / 2;

  char* ws = (char*)d_ws;
  size_t off = 0;
  auto carve = [&](size_t bytes) -> void* {
    void* p = ws + off;
    off = (off + bytes + 255) & ~(size_t)255;
    return p;
  };
  float* Hf    = (float*)carve((size_t)B_ * H_ * N_ * EMB_ * sizeof(float)); // 25 MB
  float* asrc  = (float*)carve((size_t)H_ * N_ * sizeof(float));
  float* adst  = (float*)carve((size_t)H_ * N_ * sizeof(float));
  float* seg   = (float*)carve((size_t)H_ * N_ * sizeof(float));
  int*   rp    = (int*)carve((size_t)(N_ + 1) * sizeof(int));
  float* score = (float*)carve((size_t)H_ * E * sizeof(float));
  (void)ws_size; (void)n_in; (void)out_size;

  // Stage A: fp32 WMMA GEMM per (b, h)
  gat_stageA<<<dim3(N_ / 64, H_, B_), 128, 0, stream>>>(X, Wn, bn, Hf);

  // Stage A2: per-node attention dots for batch B-1
  gat_att_coef<<<(H_ * N_ * 32) / 256, 256, 0, stream>>>(Hf, Wa, asrc, adst);

  // Zero segment sums (must be re-zeroed every call)
  gat_zero<<<(H_ * N_ + 255) / 256, 256, 0, stream>>>(seg, H_ * N_);

  // Stage B: edge scores + segment sums
  gat_edge_scores<<<(E + 255) / 256, 256, 0, stream>>>(edges, asrc, adst, ba, score, seg, E);

  // CSR row pointers (edges sorted by src)
  gat_rowptr<<<(E + 255) / 256, 256, 0, stream>>>(edges, rp, E);

  // Stage C: normalized gather-aggregate into the output
  gat_aggregate<<<(B_ * N_ * 32) / 256, 256, 0, stream>>>(edges, rp, score, seg, Hf,
                                                          (float*)d_out, E);
}